// BeeGameModule_12214886990702
// MI455X (gfx1250) — compile-verified
//
#include <hip/hip_runtime.h>
#include <hip/hip_bf16.h>

#define B_TOTAL      4096
#define NUM_AGENTS   128
#define NUM_HIVES    64
#define NUM_ENTITIES 192

typedef __attribute__((ext_vector_type(2))) float v2f;
typedef __attribute__((ext_vector_type(8))) float v8f;

// Exact fp32 wave(32)-wide sum using V_WMMA_F32_16X16X4_F32.
// A layout (16x4 f32, 2 VGPRs): VGPR0 = {lanes0-15: K=0, lanes16-31: K=2},
// VGPR1 = {K=1, K=3}. Putting v in a[0], 0 in a[1] covers all 32 lanes at
// full rank. B = ones(4x16) => D[m][n] = acc_m + acc_{m+16}.
// C/D layout: lane L holds rows (L>=16 ? 8..15 : 0..7) at col L%16, so the
// per-lane sum of the 8 C VGPRs is S_low (lanes 0-15) / S_high (lanes 16-31);
// shfl_xor(16) completes the total. EXEC must be all ones at call site.
__device__ __forceinline__ float wave_sum_f32_wmma(float v) {
    v2f a; a[0] = v;    a[1] = 0.0f;
    v2f b; b[0] = 1.0f; b[1] = 1.0f;
    v8f c = {0.f, 0.f, 0.f, 0.f, 0.f, 0.f, 0.f, 0.f};
    c = __builtin_amdgcn_wmma_f32_16x16x4_f32(
        /*neg_a=*/false, a, /*neg_b=*/false, b,
        /*c_mod=*/(short)0, c, /*reuse_a=*/false, /*reuse_b=*/false);
    float s = c[0] + c[1] + c[2] + c[3] + c[4] + c[5] + c[6] + c[7];
    s += __shfl_xor(s, 16, 32);
    return s;
}

__global__ void __launch_bounds__(NUM_AGENTS)
bee_main_kernel(const float* __restrict__ movements,    // (B,192,2)
                const float* __restrict__ votes,        // (B,128,64)
                const float* __restrict__ hive_values,  // (B,64,1)
                float* __restrict__ out,                // out[1+b] = max_freq
                float* __restrict__ partials)           // (B,) per-block cost
{
    const int b = blockIdx.x;
    const int t = threadIdx.x;

    __shared__ int   s_counts[NUM_HIVES];
    __shared__ int   s_maxc;
    __shared__ float s_mc[4];
    __shared__ float s_val[4];

    if (t < NUM_HIVES) s_counts[t] = 0;
    if (t == 0)        s_maxc = 0;
    __syncthreads();

    // ---- per-agent argmax over 64 hives (first-max-index semantics) ----
    const float4* vrow =
        (const float4*)(votes + ((size_t)b * NUM_AGENTS + t) * NUM_HIVES);
    float best = -__builtin_inff();
    int   bh   = 0;
#pragma unroll
    for (int q = 0; q < NUM_HIVES / 4; ++q) {
        float4 v = vrow[q];
        if (v.x > best) { best = v.x; bh = 4 * q + 0; }
        if (v.y > best) { best = v.y; bh = 4 * q + 1; }
        if (v.z > best) { best = v.z; bh = 4 * q + 2; }
        if (v.w > best) { best = v.w; bh = 4 * q + 3; }
    }
    atomicAdd(&s_counts[bh], 1);   // LDS integer histogram (exact)

    // ---- movement cost: 192 entities, threads cover t and 128+t (t<64) ----
    const float2* mv = (const float2*)(movements + (size_t)b * NUM_ENTITIES * 2);
    float2 m0 = mv[t];
    float  mc = sqrtf(m0.x * m0.x + m0.y * m0.y);
    if (t < NUM_ENTITIES - NUM_AGENTS) {
        float2 m1 = mv[NUM_AGENTS + t];
        mc += sqrtf(m1.x * m1.x + m1.y * m1.y);
    }

    __syncthreads();

    // value = sum_h counts[h] * hive_values[b,h]; max count via LDS atomicMax
    float valp = 0.0f;
    if (t < NUM_HIVES) {
        int c = s_counts[t];
        valp = (float)c * hive_values[(size_t)b * NUM_HIVES + t];
        atomicMax(&s_maxc, c);
    }

    // ---- wave-wide exact fp32 sums via WMMA, then cross-wave via LDS ----
    float wmc  = wave_sum_f32_wmma(mc);
    float wval = wave_sum_f32_wmma(valp);
    const int wave = t >> 5;
    if ((t & 31) == 0) { s_mc[wave] = wmc; s_val[wave] = wval; }
    __syncthreads();

    if (t == 0) {
        float mct  = s_mc[0] + s_mc[1] + s_mc[2] + s_mc[3];
        float valt = s_val[0] + s_val[1] + s_val[2] + s_val[3];
        float max_freq = (float)s_maxc * (1.0f / NUM_AGENTS);
        // discount = 100*(1 - sigmoid(30*(mf-0.7))) == 100 / (1 + exp(x))
        float x        = 30.0f * (max_freq - 0.7f);
        float discount = 100.0f / (1.0f + expf(x));
        partials[b]  = mct - valt / discount;  // movement + vote cost partial
        out[1 + b]   = max_freq;
    }
}

// Deterministic fixed-order tree reduction of the 4096 per-batch partials.
__global__ void __launch_bounds__(256)
bee_final_kernel(const float* __restrict__ partials, float* __restrict__ out)
{
    const int t = threadIdx.x;
    __shared__ float s_w[8];

    float acc = 0.0f;
#pragma unroll
    for (int i = 0; i < B_TOTAL / 256; ++i)
        acc += partials[t + i * 256];

    float ws = wave_sum_f32_wmma(acc);
    if ((t & 31) == 0) s_w[t >> 5] = ws;
    __syncthreads();

    if (t == 0) {
        float total = 0.0f;
#pragma unroll
        for (int i = 0; i < 8; ++i) total += s_w[i];
        out[0] = total;
    }
}

extern "C" void kernel_launch(void* const* d_in, const int* in_sizes, int n_in,
                              void* d_out, int out_size, void* d_ws, size_t ws_size,
                              hipStream_t stream) {
    // setup_inputs order: movements, utterances(unused), votes, hive_values,
    // locations(unused)
    const float* movements   = (const float*)d_in[0];
    const float* votes       = (const float*)d_in[2];
    const float* hive_values = (const float*)d_in[3];
    float*       out         = (float*)d_out;     // [0]=cost, [1..B]=max_freq
    float*       partials    = (float*)d_ws;      // B floats of scratch

    bee_main_kernel<<<B_TOTAL, NUM_AGENTS, 0, stream>>>(
        movements, votes, hive_values, out, partials);
    bee_final_kernel<<<1, 256, 0, stream>>>(partials, out);
}